// LevinsonDurbin_38920993636625
// MI455X (gfx1250) — compile-verified
//
#include <hip/hip_runtime.h>

// Levinson-Durbin, order N=32, T independent frames.
// pAC: [N+1, T] f32 (k-major, stride T). out: [N, T] f32.
// One thread per frame, fully unrolled -> lp[]/ac[] live entirely in VGPRs.
// T specialized as a compile-time constant (1<<20 from the reference) so the
// uniform address components (k*T*4) constant-fold into SGPR bases instead of
// per-load 64-bit VALU adds. NT hints: ~273MB stream > 192MB L2, zero reuse.

#define LPC_N 32
#define LPC_EPS 1e-5f

template <int T_CONST>
__global__ __launch_bounds__(256) void
levinson_durbin_kernel(const float* __restrict__ pAC,
                       float* __restrict__ out,
                       int T_rt) {
    const int T = (T_CONST > 0) ? T_CONST : T_rt;
    const int t = blockIdx.x * blockDim.x + threadIdx.x;
    if (t >= T) return;

    // ---- Load all 33 lags for this frame. Uniform base per k (constant when
    // T_CONST>0), divergent part is just `t` -> saddr + voffset addressing,
    // coalesced across lanes, non-temporal.
    float ac[LPC_N + 1];
#pragma unroll
    for (int k = 0; k <= LPC_N; ++k) {
        const float* base = pAC + (size_t)k * (size_t)T;   // uniform -> SGPR
        ac[k] = __builtin_nontemporal_load(base + t);
    }

    float E = ac[0];
    float lp[LPC_N];

#pragma unroll
    for (int i = 0; i < LPC_N; ++i) {
        // numerator = ac[i+1] + sum_{j<i} lp[j] * ac[i-j]
        // Two interleaved accumulators: halves the dependent FMA chain.
        float num;
        if (i == 0) {
            num = ac[1];
        } else {
            float acc0 = 0.0f, acc1 = 0.0f;
#pragma unroll
            for (int j = 0; j < i; j += 2) acc0 = fmaf(lp[j], ac[i - j], acc0);
#pragma unroll
            for (int j = 1; j < i; j += 2) acc1 = fmaf(lp[j], ac[i - j], acc1);
            num = ac[i + 1] + (acc0 + acc1);
        }

        // ki = num / E  via v_rcp_f32 + one Newton-Raphson step
        // (avoids the ~10-op precise-division expansion, x32 iterations).
        float inv = __builtin_amdgcn_rcpf(E);
        inv = fmaf(fmaf(-E, inv, 1.0f), inv, inv);
        const float ki = num * inv;

        // E *= clip(1 - ki*ki, EPS, inf)
        const float c = fmaxf(fmaf(-ki, ki, 1.0f), LPC_EPS);
        E *= c;

        // lp[:i] = lp[:i] - ki * lp[:i][::-1]   (from OLD lp)
        // Pairwise symmetric update: 2 independent FMAs per pair, no temps.
        if (i > 0) {
#pragma unroll
            for (int j = 0; j < i / 2; ++j) {
                const float lo = lp[j];
                const float hi = lp[i - 1 - j];
                lp[j]         = fmaf(-ki, hi, lo);
                lp[i - 1 - j] = fmaf(-ki, lo, hi);
            }
            if (i & 1) {
                const int m = i / 2;
                lp[m] = fmaf(-ki, lp[m], lp[m]);   // lp[m] * (1 - ki)
            }
        }

        lp[i] = -ki;
    }

    // ---- Store (uniform SGPR base per row, coalesced, NT).
#pragma unroll
    for (int j = 0; j < LPC_N; ++j) {
        float* base = out + (size_t)j * (size_t)T;
        __builtin_nontemporal_store(lp[j], base + t);
    }
}

extern "C" void kernel_launch(void* const* d_in, const int* in_sizes, int n_in,
                              void* d_out, int out_size, void* d_ws, size_t ws_size,
                              hipStream_t stream) {
    (void)n_in; (void)d_ws; (void)ws_size; (void)out_size;
    const float* pAC = (const float*)d_in[0];
    float* out = (float*)d_out;

    const int T = in_sizes[0] / (LPC_N + 1);   // pAC has (N+1)*T elements

    const int block = 256;                     // 8 wave32 per workgroup
    const int grid = (T + block - 1) / block;

    if (T == (1 << 20)) {
        // Specialized: all uniform address offsets are compile-time constants.
        levinson_durbin_kernel<(1 << 20)><<<grid, block, 0, stream>>>(pAC, out, T);
    } else {
        levinson_durbin_kernel<0><<<grid, block, 0, stream>>>(pAC, out, T);
    }
}